// multi_shallow_embedding_with_static_45174466019579
// MI455X (gfx1250) — compile-verified
//
#include <hip/hip_runtime.h>
#include <cstdint>

typedef __attribute__((ext_vector_type(2))) float v2f;
typedef __attribute__((ext_vector_type(8))) float v8f;

#define NGRAPH 11
#define WS_STRIDE 512   // uints per graph: [0..255]=hist, 256=prefix, 257=pmask, 258=kk, 259=ties
#define MAXN 2048

struct GMeta { const float* s; const float* t; float* out; int N; int pad; };
struct Metas { GMeta m[NGRAPH]; };

// Order-preserving map: f32 -> u32 (larger float => larger uint)
__device__ __forceinline__ unsigned map_f32(float f) {
  unsigned b = __float_as_uint(f);
  return (b & 0x80000000u) ? ~b : (b | 0x80000000u);
}

// One 16x16 tile of the rank-1 outer product s[r0+M]*t[c0+N] via f32 WMMA.
// A (16x4, MxK): lanes 0-15 hold M=lane with VGPR0=K0,VGPR1=K1; lanes 16-31 hold K2,K3.
// B (4x16, KxN): lanes 0-15 hold N=lane with VGPR0=K0,VGPR1=K1; lanes 16-31 hold K2,K3.
// Only K=0 is populated -> D[M][N] = s[r0+M]*t[c0+N] exactly (fma with 0 acc).
__device__ __forceinline__ v8f outer_tile(const float* ss, const float* tt,
                                          int r0, int c0, int lane) {
  v2f a; a.x = 0.f; a.y = 0.f;
  v2f b; b.x = 0.f; b.y = 0.f;
  if (lane < 16) {
    a.x = ss[r0 + lane];
    b.x = tt[c0 + lane];
  }
  v8f c = {};
  c = __builtin_amdgcn_wmma_f32_16x16x4_f32(false, a, false, b, (short)0, c, false, false);
  return c;
}

__global__ void init_kernel(unsigned* ws, const int* kptr) {
  unsigned* base = ws + (size_t)blockIdx.x * WS_STRIDE;
  base[threadIdx.x] = 0u;                 // zero the 256 histogram bins
  if (threadIdx.x == 0) {
    base[256] = 0u;                       // prefix
    base[257] = 0u;                       // prefix mask
    base[258] = (unsigned)(*kptr);        // remaining rank kk
    base[259] = 0u;                       // tie ticket counter
  }
}

__global__ void __launch_bounds__(256) hist_kernel(Metas metas, unsigned* ws, int pass) {
  __shared__ unsigned lh[8][256];         // per-wave private histograms
  __shared__ float ss[MAXN];
  __shared__ float tt[MAXN];
  const int g = blockIdx.y;
  GMeta m = metas.m[g];
  unsigned* hist = ws + (size_t)g * WS_STRIDE;
  const unsigned prefix = hist[256];
  const unsigned pmask  = hist[257];
  const int tid = threadIdx.x;
  for (int i = tid; i < 8 * 256; i += 256) (&lh[0][0])[i] = 0u;
  const int N = m.N;
  for (int i = tid; i < N; i += 256) { ss[i] = m.s[i]; tt[i] = m.t[i]; }
  __syncthreads();

  const int tilesX = N >> 4;
  const int nTiles = tilesX * tilesX;
  const int wave = tid >> 5;
  const int lane = tid & 31;
  const int shift = 24 - 8 * pass;
  unsigned* wh = lh[wave];

  for (int tile = blockIdx.x * 8 + wave; tile < nTiles; tile += (int)gridDim.x * 8) {
    const int tm = tile / tilesX;
    const int tn = tile - tm * tilesX;
    v8f c = outer_tile(ss, tt, tm << 4, tn << 4, lane);
    const int col = (tn << 4) + (lane & 15);
    const int mb  = (tm << 4) + ((lane >> 4) << 3);
#pragma unroll
    for (int r = 0; r < 8; ++r) {
      unsigned u = map_f32(c[r]);
      if (mb + r == col) u = 0u;          // masked diagonal: below any real value's map
      if ((u & pmask) == prefix)
        atomicAdd(&wh[(u >> shift) & 255u], 1u);
    }
  }
  __syncthreads();
  unsigned sum = 0;
#pragma unroll
  for (int w = 0; w < 8; ++w) sum += lh[w][tid];
  if (sum) atomicAdd(&hist[tid], sum);
}

__global__ void scan_kernel(unsigned* ws, int pass) {
  unsigned* hist = ws + (size_t)blockIdx.x * WS_STRIDE;
  if (threadIdx.x == 0) {
    unsigned kk = hist[258];
    unsigned cum = 0, chosen = 0;
    for (int b = 255; b >= 0; --b) {
      const unsigned h = hist[b];
      if (cum + h >= kk) { chosen = (unsigned)b; kk -= cum; break; }
      cum += h;
    }
    const int shift = 24 - 8 * pass;
    hist[256] |= chosen << shift;
    hist[257] |= 0xFFu << shift;
    hist[258] = kk;                       // rank remaining within chosen bin
  }
  __syncthreads();                        // thread0 done reading before zeroing
  hist[threadIdx.x] = 0u;                 // reset bins for next pass
}

__global__ void __launch_bounds__(256) write_kernel(Metas metas, unsigned* ws) {
  __shared__ float ss[MAXN];
  __shared__ float tt[MAXN];
  const int g = blockIdx.y;
  GMeta m = metas.m[g];
  unsigned* hist = ws + (size_t)g * WS_STRIDE;
  const unsigned T  = hist[256];          // exact 32-bit threshold (k-th largest)
  const unsigned kk = hist[258];          // how many elements equal to T to accept
  const int tid = threadIdx.x;
  const int N = m.N;
  for (int i = tid; i < N; i += 256) { ss[i] = m.s[i]; tt[i] = m.t[i]; }
  __syncthreads();

  const int tilesX = N >> 4;
  const int nTiles = tilesX * tilesX;
  const int wave = tid >> 5;
  const int lane = tid & 31;

  for (int tile = blockIdx.x * 8 + wave; tile < nTiles; tile += (int)gridDim.x * 8) {
    const int tm = tile / tilesX;
    const int tn = tile - tm * tilesX;
    v8f c = outer_tile(ss, tt, tm << 4, tn << 4, lane);
    const int col = (tn << 4) + (lane & 15);
    const int mb  = (tm << 4) + ((lane >> 4) << 3);
#pragma unroll
    for (int r = 0; r < 8; ++r) {
      const int row = mb + r;
      unsigned u = map_f32(c[r]);
      if (row == col) u = 0u;
      float o = 0.f;
      if (u > T) {
        o = 1.f;
      } else if (u == T) {
        const unsigned ticket = atomicAdd(&hist[259], 1u);
        if (ticket < kk) o = 1.f;
      }
      m.out[(size_t)row * (size_t)N + (size_t)col] = o;
    }
  }
}

extern "C" void kernel_launch(void* const* d_in, const int* in_sizes, int n_in,
                              void* d_out, int out_size, void* d_ws, size_t ws_size,
                              hipStream_t stream) {
  const float* sdyn = (const float*)d_in[0];
  const float* tdyn = (const float*)d_in[1];
  const float* sst  = (const float*)d_in[2];
  const float* tst  = (const float*)d_in[3];
  const float* sic  = (const float*)d_in[4];
  const float* tic  = (const float*)d_in[5];
  const float* srp  = (const float*)d_in[6];
  const float* trp  = (const float*)d_in[7];
  const int*   kptr = (const int*)d_in[8];
  float* out = (float*)d_out;

  const int G  = 8;
  const int Nd = in_sizes[0] / G;   // 2048
  const int Ns = in_sizes[2];       // 512
  const int Ni = in_sizes[4];       // 2048
  const int Nr = in_sizes[6];       // 1024

  Metas metas;
  size_t off = 0;
  for (int g = 0; g < G; ++g) {
    metas.m[g].s = sdyn + (size_t)g * Nd;
    metas.m[g].t = tdyn + (size_t)g * Nd;
    metas.m[g].out = out + off;
    metas.m[g].N = Nd; metas.m[g].pad = 0;
    off += (size_t)Nd * Nd;
  }
  metas.m[8]  = { sst, tst, out + off, Ns, 0 }; off += (size_t)Ns * Ns;
  metas.m[9]  = { sic, tic, out + off, Ni, 0 }; off += (size_t)Ni * Ni;
  metas.m[10] = { srp, trp, out + off, Nr, 0 };

  unsigned* ws = (unsigned*)d_ws;

  init_kernel<<<dim3(NGRAPH), dim3(256), 0, stream>>>(ws, kptr);

  dim3 grid(128, NGRAPH);
  for (int pass = 0; pass < 4; ++pass) {
    hist_kernel<<<grid, dim3(256), 0, stream>>>(metas, ws, pass);
    scan_kernel<<<dim3(NGRAPH), dim3(256), 0, stream>>>(ws, pass);
  }
  write_kernel<<<grid, dim3(256), 0, stream>>>(metas, ws);
}